// Discriminator_38946763441062
// MI455X (gfx1250) — compile-verified
//
#include <hip/hip_runtime.h>
#include <math.h>

// Problem constants (match reference)
#define NUM_LAYERS 15
#define H          100
#define SEQ        16384
#define CH         128              // timesteps per pipeline chunk (8 waves x 16 rows)
#define NC         (SEQ / CH)       // 128 chunks
#define NSTR       112              // padded N (7 x 16) for W_ih^T / pre
#define WSTR       102              // even stride for W_hh -> aligned b64 rows
#define NT         7                // N tiles of 16

typedef __attribute__((ext_vector_type(2))) float v2f;
typedef __attribute__((ext_vector_type(8))) float v8f;
typedef __attribute__((ext_vector_type(4))) unsigned int u32x4;
typedef __attribute__((ext_vector_type(4))) int i32x4;
typedef __attribute__((ext_vector_type(8))) int i32x8;

// Dynamic LDS partition (float offsets)
#define OFF_WTP  0                          // W_ih^T K-pair interleaved: [k/2][n][2], 50 x 224
#define OFF_WHH  (OFF_WTP + 50 * (2*NSTR))  // W_hh [o][i], 100 x 102
#define OFF_PRE  (OFF_WHH + H * WSTR)       // pre/h [t][n], 128 x 112
#define OFF_XB   (OFF_PRE + CH * NSTR)      // TDM-staged input tile, 128 x 100
#define OFF_BS   (OFF_XB  + CH * H)         // b_ih + b_hh, padded to 112
#define OFF_HB   (OFF_BS  + NSTR)           // hidden state, 128
#define OFF_W7   (OFF_HB  + 128)            // final linear weights
#define OFF_B7   (OFF_W7  + H)
#define SMEM_FLOATS (OFF_B7 + 4)            // 48880 floats = 195,520 B (< 320KB/WGP)

__global__ void rnn_init(int* flags, int n) {
    int i = blockIdx.x * blockDim.x + threadIdx.x;
    if (i < n) flags[i] = 0;
}

// Issue one TDM 2D tile load (rows x 100 f32, row stride 100) into LDS offset.
__device__ __forceinline__ void tdm_load_tile(const float* gsrc, unsigned lds_byte_off) {
    unsigned long long ga = (unsigned long long)(uintptr_t)gsrc;
    u32x4 g0;
    g0[0] = 1u;                                        // count=1 (valid descriptor)
    g0[1] = lds_byte_off;                              // lds_addr (bytes)
    g0[2] = (unsigned)(ga & 0xFFFFFFFFu);              // global_addr[31:0]
    g0[3] = (unsigned)((ga >> 32) & 0x1FFFFFFu) | 0x80000000u; // addr[56:32] | type=2
    i32x8 g1;
    g1[0] = 0x20000;                                   // data_size=2 (4B), no multicast
    g1[1] = (int)(100u << 16);                         // tensor_dim0 = 100 (low16 at bit48)
    g1[2] = (int)(16384u << 16);                       // tensor_dim1 = SEQ rows (low16 at bit80)
    g1[3] = (int)(100u << 16);                         // tile_dim0 = 100
    g1[4] = CH;                                        // tile_dim1 = 128 rows
    g1[5] = 100;                                       // tensor_dim0_stride = 100 elements
    g1[6] = 0;
    g1[7] = 0;
    i32x4 z4 = {};
#if defined(__clang_major__) && __clang_major__ >= 23
    i32x8 z8 = {};
    __builtin_amdgcn_tensor_load_to_lds(g0, g1, z4, z4, z8, 0);
#else
    __builtin_amdgcn_tensor_load_to_lds(g0, g1, z4, z4, 0);
#endif
    __builtin_amdgcn_s_wait_tensorcnt(0);
}

__global__ __launch_bounds__(256, 1)
void rnn_pipeline(const float* __restrict__ x_in,
                  const float* __restrict__ W_ih,
                  const float* __restrict__ W_hh,
                  const float* __restrict__ b_ih,
                  const float* __restrict__ b_hh,
                  const float* __restrict__ h0,
                  const float* __restrict__ W7,
                  const float* __restrict__ b7,
                  float* __restrict__ out,
                  float* __restrict__ buf0,
                  float* __restrict__ buf1,
                  int*   __restrict__ flags)
{
    extern __shared__ float sm[];
    float* wtp = sm + OFF_WTP;
    float* whh = sm + OFF_WHH;
    float* pre = sm + OFF_PRE;
    float* xb  = sm + OFF_XB;
    float* bs  = sm + OFF_BS;
    float* hb  = sm + OFF_HB;
    float* w7s = sm + OFF_W7;
    float* b7s = sm + OFF_B7;

    const int l     = blockIdx.x;          // one workgroup per layer (pipeline stage)
    const int tid   = threadIdx.x;
    const int wv    = tid >> 5;            // wave -> M-tile (16 timesteps)
    const int lane  = tid & 31;
    const int mrow  = lane & 15;           // M row (A) / N column (B,C) within tile
    const int khalf = (lane >> 4) << 1;    // lanes 16-31 handle K+2,K+3
    const int mhi   = (lane >> 4) << 3;    // C layout: upper lanes hold M+8
    const int srow  = tid >> 1;            // scan: 2 threads per row
    const int shalf = tid & 1;             // scan: column half (0..49 / 50..99)

    const float* Xprev = (l == 0) ? x_in : ((l & 1) ? buf0 : buf1);
    float*       Xout  = (l & 1) ? buf1 : buf0;

    // ---- Stage layer weights into LDS (re-read 16384x per layer -> must be LDS) ----
    for (int i = tid; i < 50 * (2*NSTR); i += 256) wtp[i] = 0.f;
    __syncthreads();
    for (int i = tid; i < H * H; i += 256) {
        int o = i / H, k = i % H;
        // K-pair interleaved W_ih^T: lane's B pair {WT[k][n],WT[k+1][n]} contiguous -> 1x ds_load_b64
        wtp[(k >> 1) * (2*NSTR) + o * 2 + (k & 1)] = W_ih[(size_t)l * H * H + i];
    }
    for (int i = tid; i < H * H; i += 256) {
        int o = i / H, k = i % H;
        whh[o * WSTR + k] = W_hh[(size_t)l * H * H + i];
    }
    for (int i = tid; i < NSTR; i += 256)
        bs[i] = (i < H) ? (b_ih[l * H + i] + b_hh[l * H + i]) : 0.f;
    if (tid < H) hb[tid] = h0[l * H + tid];
    if (l == NUM_LAYERS - 1) {
        if (tid < H)  w7s[tid] = W7[tid];
        if (tid == 0) b7s[0]   = b7[0];
    }
    __syncthreads();

    for (int c = 0; c < NC; ++c) {
        // ---- wait for producer layer's chunk (device-scope acquire) ----
        if (l > 0) {
            if (tid == 0) {
                while (__hip_atomic_load(&flags[(l - 1) * NC + c],
                                         __ATOMIC_ACQUIRE,
                                         __HIP_MEMORY_SCOPE_AGENT) == 0) {
                    __builtin_amdgcn_s_sleep(1);
                }
            }
            __syncthreads();
        }

        // ---- TDM: stage this chunk's input tile (128 x 100 f32) into LDS ----
        if (wv == 0) {
            tdm_load_tile(Xprev + (size_t)c * CH * H, (unsigned)(OFF_XB * sizeof(float)));
        }
        __syncthreads();

        // ---- Input projection GEMM: pre = X_chunk @ W_ih^T  (WMMA f32 16x16x4) ----
        const float* arow = xb + (wv * 16 + mrow) * H;

        v8f acc[NT];
        #pragma unroll
        for (int nt = 0; nt < NT; ++nt) { v8f z = {}; acc[nt] = z; }

        for (int k0 = 0; k0 < H; k0 += 4) {
            const int kk = k0 + khalf;
            v2f a = *reinterpret_cast<const v2f*>(arow + kk);       // A: M=mrow, K=kk,kk+1
            const float* bp = wtp + (kk >> 1) * (2*NSTR) + mrow * 2;
            #pragma unroll
            for (int nt = 0; nt < NT; ++nt) {
                v2f b = *reinterpret_cast<const v2f*>(bp + nt * 32); // B: N=mrow, K=kk,kk+1
                acc[nt] = __builtin_amdgcn_wmma_f32_16x16x4_f32(
                    false, a, false, b, (short)0, acc[nt], false, false);
            }
        }

        // ---- accumulators -> LDS pre (+ fused bias) ----
        #pragma unroll
        for (int nt = 0; nt < NT; ++nt) {
            #pragma unroll
            for (int r = 0; r < 8; ++r) {
                int m = wv * 16 + mhi + r;
                pre[m * NSTR + nt * 16 + mrow] = acc[nt][r] + bs[nt * 16 + mrow];
            }
        }
        __syncthreads();

        // ---- Sequential recurrence: h_t = tanh(pre_t + W_hh h_{t-1}) ----
        // 2 threads per row, 25 float2 FMAs each (2 accumulators), combine via shfl_xor.
        for (int t = 0; t < CH; ++t) {
            float hnew = 0.f;
            if (srow < H) {
                const float* wr = whh + srow * WSTR + shalf * 50;
                const float* hr = hb + shalf * 50;
                v2f sa = {}, sb = {};
                #pragma unroll
                for (int i = 0; i < 25; i += 2) {
                    sa += *reinterpret_cast<const v2f*>(wr + 2*i) *
                          *reinterpret_cast<const v2f*>(hr + 2*i);
                    if (i + 1 < 25)
                        sb += *reinterpret_cast<const v2f*>(wr + 2*i + 2) *
                              *reinterpret_cast<const v2f*>(hr + 2*i + 2);
                }
                float s = sa.x + sa.y + sb.x + sb.y;
                s += __shfl_xor(s, 1, 32);               // add partner half's sum
                hnew = tanhf(s + pre[t * NSTR + srow]);
            }
            __syncthreads();
            if (srow < H && shalf == 0) {
                hb[srow] = hnew;
                pre[t * NSTR + srow] = hnew;             // stash h_t for copy-out / head
            }
            __syncthreads();
        }

        // ---- emit chunk ----
        if (l == NUM_LAYERS - 1) {
            if (tid < CH) {                              // fused 100->1 linear + sigmoid
                float s = b7s[0];
                #pragma unroll 4
                for (int cc = 0; cc < H; ++cc) s += pre[tid * NSTR + cc] * w7s[cc];
                out[c * CH + tid] = 1.0f / (1.0f + __expf(-s));
            }
        } else {
            for (int i = tid; i < CH * H; i += 256) {
                int tt = i / H, n = i % H;
                Xout[(size_t)(c * CH + tt) * H + n] = pre[tt * NSTR + n];
            }
            __threadfence();
            __syncthreads();
            if (tid == 0)
                __hip_atomic_store(&flags[l * NC + c], 1, __ATOMIC_RELEASE,
                                   __HIP_MEMORY_SCOPE_AGENT);
        }
        __syncthreads();   // protect pre/xb before next chunk overwrites them
    }
}

extern "C" void kernel_launch(void* const* d_in, const int* in_sizes, int n_in,
                              void* d_out, int out_size, void* d_ws, size_t ws_size,
                              hipStream_t stream) {
    const float* x   = (const float*)d_in[0];
    const float* Wih = (const float*)d_in[1];
    const float* Whh = (const float*)d_in[2];
    const float* bih = (const float*)d_in[3];
    const float* bhh = (const float*)d_in[4];
    const float* h0  = (const float*)d_in[5];
    const float* W7  = (const float*)d_in[6];
    const float* b7  = (const float*)d_in[7];
    float* out = (float*)d_out;

    int* flags = (int*)d_ws;
    size_t flag_bytes = ((size_t)NUM_LAYERS * NC * sizeof(int) + 255) & ~(size_t)255;
    float* buf0 = (float*)((char*)d_ws + flag_bytes);
    float* buf1 = buf0 + (size_t)SEQ * H;

    rnn_init<<<(NUM_LAYERS * NC + 255) / 256, 256, 0, stream>>>(flags, NUM_LAYERS * NC);
    rnn_pipeline<<<NUM_LAYERS, 256, SMEM_FLOATS * sizeof(float), stream>>>(
        x, Wih, Whh, bih, bhh, h0, W7, b7, out, buf0, buf1, flags);
}